// RoiPoolingConv_2748779069995
// MI455X (gfx1250) — compile-verified
//
#include <hip/hip_runtime.h>
#include <hip/hip_bf16.h>

// ROI pooling (crop + nearest resize), MI455X / gfx1250.
//
// img : [1, 512, 100, 100] f32
// rois: [1, R, 4] i32 (x, y, w, h), in-bounds by construction
// out : flat [R, 512, 14, 14] f32 (the [1,R,14,14,512] view is a raw
//       reinterpretation of the same contiguous buffer)
//
// Strategy (write-bandwidth-bound kernel, ~401 MB out, 20.5 MB image fits L2):
//  - one block per (ROI, 64-channel group)
//  - 196 source offsets computed once per block into LDS (shared by all
//    channels of the ROI) -> ds_load_b128 per thread-iteration
//  - gathers via 4x global_load_b32 (default temporal hint: keep image in L2)
//  - output via aligned float4 NON-TEMPORAL stores so the 401 MB write
//    stream does not evict the L2-resident image

#define POOL   14
#define PP     (POOL * POOL)      // 196
#define NQ     (PP / 4)           // 49 float4 per channel (196 % 4 == 0)
#define CCH    512
#define IMG_H  100
#define IMG_W  100
#define IMG_HW (IMG_H * IMG_W)
#define CHUNK  64                 // channels per block
#define NGRP   (CCH / CHUNK)      // 8 channel groups
#define BLOCK  256                // 8 wave32

typedef float v4f __attribute__((ext_vector_type(4)));

__global__ __launch_bounds__(BLOCK) void roi_pool_nt_kernel(
    const float* __restrict__ img,
    const int*   __restrict__ rois,
    float*       __restrict__ out)
{
    __shared__ __attribute__((aligned(16))) int offs[PP];

    const int r  = blockIdx.x;           // ROI index
    const int cg = blockIdx.y;           // channel group 0..7

    // ROI descriptor: uniform per block -> compiler emits scalar loads.
    const int x = rois[r * 4 + 0];
    const int y = rois[r * 4 + 1];
    const int w = rois[r * 4 + 2];
    const int h = rois[r * 4 + 3];

    const int t = threadIdx.x;

    // Build the 196-entry source-offset table once per block.
    if (t < PP) {
        const int py  = t / POOL;
        const int px  = t - py * POOL;          // t % 14
        const int row = y + (py * h) / POOL;    // nearest-neighbor source row
        const int col = x + (px * w) / POOL;    // nearest-neighbor source col
        offs[t] = row * IMG_W + col;
    }
    __syncthreads();

    const float* fbase = img + (size_t)(cg * CHUNK) * IMG_HW;
    float*       obase = out + (size_t)r * (CCH * PP)
                             + (size_t)(cg * CHUNK) * PP;

    // CHUNK*NQ = 3136 float4-stores per block; 256 threads -> 13 iterations.
    #pragma unroll 2
    for (int i = t; i < CHUNK * NQ; i += BLOCK) {
        const int c = i / NQ;                   // 0..63 (const-div -> mul/shift)
        const int q = i - c * NQ;               // 0..48

        // 16B-aligned LDS read of 4 consecutive offsets (ds_load_b128).
        const int4 o4 = *reinterpret_cast<const int4*>(&offs[q * 4]);

        const float* fp = fbase + (size_t)c * IMG_HW;
        v4f v;
        v.x = fp[o4.x];
        v.y = fp[o4.y];
        v.z = fp[o4.z];
        v.w = fp[o4.w];

        // 16B-aligned non-temporal store: obase + c*784B + q*16B.
        v4f* dst = reinterpret_cast<v4f*>(obase + (size_t)c * PP + q * 4);
        __builtin_nontemporal_store(v, dst);
    }
}

extern "C" void kernel_launch(void* const* d_in, const int* in_sizes, int n_in,
                              void* d_out, int out_size, void* d_ws, size_t ws_size,
                              hipStream_t stream) {
    const float* img  = (const float*)d_in[0];
    const int*   rois = (const int*)d_in[1];
    float*       out  = (float*)d_out;

    const int R = in_sizes[1] / 4;   // 1000 in the reference

    dim3 grid(R, NGRP, 1);
    dim3 block(BLOCK, 1, 1);
    roi_pool_nt_kernel<<<grid, block, 0, stream>>>(img, rois, out);
}